// FullAdaptiveModel_979252544297
// MI455X (gfx1250) — compile-verified
//
#include <hip/hip_runtime.h>
#include <hip/hip_bf16.h>

#define BB    8
#define LL    1024
#define HH    256
#define HH2   512
#define HALF_ 128
#define VV    50257
#define ALPHA_ 0.9f
#define GT2    0.16f      // GATE_THRESH^2
#define LN_EPS_ 1e-5f
#define NORM_EPS_ 1e-12f

typedef __attribute__((ext_vector_type(2))) float v2f;
typedef __attribute__((ext_vector_type(8))) float v8f;

// Issue a 16-byte global->LDS async copy (ASYNCcnt path, gfx1250).
// loff: wave-relative LDS byte offset; goff: 32-bit byte offset from base.
__device__ __forceinline__ void async_g2l_b128(unsigned loff, unsigned goff,
                                               const void* base) {
    asm volatile("global_load_async_to_lds_b128 %0, %1, %2"
                 :: "v"(loff), "v"(goff), "s"(base) : "memory");
}
__device__ __forceinline__ void wait_async0() {
    asm volatile("s_wait_asynccnt 0x0" ::: "memory");
}

// ---------------------------------------------------------------------------
// Kernel 1: h = embed[seq]; x = h + MLP(h); h_ln = LayerNorm(x)
// One block = 16 tokens, 256 threads (8 waves). WMMA f32 16x16x4.
// ---------------------------------------------------------------------------
__global__ __launch_bounds__(256) void k_embed_ff_ln(
    const int* __restrict__ seq, const float* __restrict__ embed,
    const float* __restrict__ W1, const float* __restrict__ b1,
    const float* __restrict__ W2, const float* __restrict__ b2,
    const float* __restrict__ gamma, const float* __restrict__ beta,
    float* __restrict__ h_ln)
{
    __shared__ float sh_h[16][HH + 4];    // token tile, padded (260%64==4)
    __shared__ float sh_a[16][HH2 + 4];   // relu(hW1+b1), padded (516%64==4)

    const int blk  = blockIdx.x;          // 0..511
    const int b    = blk >> 6;            // L/16 = 64 tiles per batch
    const int tile = blk & 63;
    const int tid  = threadIdx.x;
    const int wave = tid >> 5;
    const int lane = tid & 31;
    const int mrow = lane & 15;
    const bool hi  = lane >= 16;

    // gather h tile: async global->LDS (16B per lane per issue)
    {
        const unsigned lds_h = (unsigned)(uintptr_t)&sh_h[0][0];
        for (int i = tid; i < 16 * (HH / 4); i += 256) {
            const int row = i >> 6, c4 = i & 63;
            const int tok = seq[(size_t)b * LL + tile * 16 + row];
            const unsigned goff = (unsigned)tok * (HH * 4u) + (unsigned)c4 * 16u;
            const unsigned loff = lds_h + ((unsigned)row * (HH + 4) + (unsigned)c4 * 4) * 4u;
            async_g2l_b128(loff, goff, embed);
        }
        wait_async0();
    }
    __syncthreads();

    // FF1: a = relu(h @ W1 + b1), 32 N-tiles over 8 waves -> 4 each
    for (int ntl = 0; ntl < 4; ++ntl) {
        const int nt = wave * 4 + ntl;
        const int n  = nt * 16 + mrow;
        v8f acc = {};
        #pragma unroll 4
        for (int k0 = 0; k0 < HH; k0 += 4) {
            const int ka = k0 + (hi ? 2 : 0);
            v2f a;  a[0]  = sh_h[mrow][ka];
                    a[1]  = sh_h[mrow][ka + 1];
            v2f bf; bf[0] = W1[(size_t)ka * HH2 + n];
                    bf[1] = W1[(size_t)(ka + 1) * HH2 + n];
            acc = __builtin_amdgcn_wmma_f32_16x16x4_f32(
                false, a, false, bf, (short)0, acc, false, false);
        }
        const float bias = b1[n];
        #pragma unroll
        for (int v = 0; v < 8; ++v) {
            const int row = v + (hi ? 8 : 0);
            const float val = acc[v] + bias;
            sh_a[row][n] = val > 0.f ? val : 0.f;
        }
    }
    __syncthreads();

    // FF2: x = h + a @ W2 + b2  (write x back in place of h)
    for (int ntl = 0; ntl < 2; ++ntl) {
        const int nt = wave * 2 + ntl;
        const int n  = nt * 16 + mrow;
        v8f acc = {};
        #pragma unroll 4
        for (int k0 = 0; k0 < HH2; k0 += 4) {
            const int ka = k0 + (hi ? 2 : 0);
            v2f a;  a[0]  = sh_a[mrow][ka];
                    a[1]  = sh_a[mrow][ka + 1];
            v2f bf; bf[0] = W2[(size_t)ka * HH + n];
                    bf[1] = W2[(size_t)(ka + 1) * HH + n];
            acc = __builtin_amdgcn_wmma_f32_16x16x4_f32(
                false, a, false, bf, (short)0, acc, false, false);
        }
        const float bias = b2[n];
        #pragma unroll
        for (int v = 0; v < 8; ++v) {
            const int row = v + (hi ? 8 : 0);
            sh_h[row][n] = sh_h[row][n] + acc[v] + bias;   // x
        }
    }
    __syncthreads();

    // LayerNorm: 8 waves x 2 rows, wave32 shuffle reduction
    for (int rr = 0; rr < 2; ++rr) {
        const int row = wave * 2 + rr;
        float s = 0.f, ss = 0.f;
        for (int c = lane; c < HH; c += 32) {
            const float x = sh_h[row][c];
            s += x; ss += x * x;
        }
        #pragma unroll
        for (int off = 16; off > 0; off >>= 1) {
            s  += __shfl_xor(s,  off, 32);
            ss += __shfl_xor(ss, off, 32);
        }
        const float mu  = s * (1.f / HH);
        const float var = ss * (1.f / HH) - mu * mu;
        const float inv = rsqrtf(var + LN_EPS_);
        float* dst = h_ln + ((size_t)b * LL + tile * 16 + row) * HH;
        for (int c = lane; c < HH; c += 32)
            dst[c] = (sh_h[row][c] - mu) * inv * gamma[c] + beta[c];
    }
}

// ---------------------------------------------------------------------------
// Kernel 2: ks_all = h_ln @ Wsem ; ke_all = h_ln @ Wepi  (all L tokens;
// token L-1 doubles as q@Wsem / q@Wepi for the readout)
// ---------------------------------------------------------------------------
__global__ __launch_bounds__(256) void k_proj(
    const float* __restrict__ h_ln,
    const float* __restrict__ Wsem, const float* __restrict__ Wepi,
    float* __restrict__ ks_all, float* __restrict__ ke_all)
{
    __shared__ float sh_h[16][HH + 4];

    const int blk  = blockIdx.x;
    const int b    = blk >> 6;
    const int tile = blk & 63;
    const int tid  = threadIdx.x;
    const int wave = tid >> 5;
    const int lane = tid & 31;
    const int mrow = lane & 15;
    const bool hi  = lane >= 16;

    // stage h_ln tile: async global->LDS
    {
        const unsigned lds_h = (unsigned)(uintptr_t)&sh_h[0][0];
        const unsigned tbase = ((unsigned)b * LL + (unsigned)tile * 16) * (HH * 4u);
        for (int i = tid; i < 16 * (HH / 4); i += 256) {
            const int row = i >> 6, c4 = i & 63;
            const unsigned goff = tbase + (unsigned)row * (HH * 4u) + (unsigned)c4 * 16u;
            const unsigned loff = lds_h + ((unsigned)row * (HH + 4) + (unsigned)c4 * 4) * 4u;
            async_g2l_b128(loff, goff, h_ln);
        }
        wait_async0();
    }
    __syncthreads();

    // 8 N-tiles of 16 over HALF=128 -> one per wave; do Wsem and Wepi
    const int n = wave * 16 + mrow;
    v8f accS = {}, accE = {};
    #pragma unroll 4
    for (int k0 = 0; k0 < HH; k0 += 4) {
        const int ka = k0 + (hi ? 2 : 0);
        v2f a;  a[0]  = sh_h[mrow][ka];
                a[1]  = sh_h[mrow][ka + 1];
        v2f bs; bs[0] = Wsem[(size_t)ka * HALF_ + n];
                bs[1] = Wsem[(size_t)(ka + 1) * HALF_ + n];
        v2f be; be[0] = Wepi[(size_t)ka * HALF_ + n];
                be[1] = Wepi[(size_t)(ka + 1) * HALF_ + n];
        accS = __builtin_amdgcn_wmma_f32_16x16x4_f32(
            false, a, false, bs, (short)0, accS, false, false);
        accE = __builtin_amdgcn_wmma_f32_16x16x4_f32(
            false, a, false, be, (short)0, accE, false, false);
    }
    #pragma unroll
    for (int v = 0; v < 8; ++v) {
        const int row = v + (hi ? 8 : 0);
        const size_t t = (size_t)b * LL + tile * 16 + row;
        ks_all[t * HALF_ + n] = accS[v];
        ke_all[t * HALF_ + n] = accE[v];
    }
}

// ---------------------------------------------------------------------------
// Kernel 3: sequential scan. One block per batch; Ms/Me live in LDS
// (2 * 128*129*4 ~= 132 KB -- needs CDNA5's 320 KB WGP LDS).
// ---------------------------------------------------------------------------
__global__ __launch_bounds__(512) void k_scan(
    const float* __restrict__ ks_all, const float* __restrict__ ke_all,
    float* __restrict__ c_out)
{
    __shared__ float Ms[HALF_][HALF_ + 1];    // padded: conflict-free matvec
    __shared__ float Me[HALF_][HALF_ + 1];
    __shared__ float sh_ks[HALF_], sh_ke[HALF_];
    __shared__ float sh_rs[HALF_], sh_re[HALF_];
    __shared__ float sh_part[4][HALF_];
    __shared__ float sh_sc[4];                // |ks|^2 |ke|^2 |rs|^2 |re|^2

    const int b    = blockIdx.x;
    const int tid  = threadIdx.x;
    const int lane = tid & 31;
    const int wave = tid >> 5;

    for (int i = tid; i < HALF_ * (HALF_ + 1); i += 512) {
        (&Ms[0][0])[i] = 0.f;
        (&Me[0][0])[i] = 0.f;
    }
    __syncthreads();

    for (int t = 0; t < LL - 1; ++t) {
        if (tid < HALF_) {
            const size_t o = ((size_t)b * LL + t) * HALF_;
            sh_ks[tid] = ks_all[o + tid];
            sh_ke[tid] = ke_all[o + tid];
        }
        __syncthreads();

        if (wave < 2) {                       // key norms (wave32 reduce)
            const float* src = wave == 0 ? sh_ks : sh_ke;
            float s = 0.f;
            #pragma unroll
            for (int i = lane; i < HALF_; i += 32) { const float x = src[i]; s += x * x; }
            #pragma unroll
            for (int off = 16; off > 0; off >>= 1) s += __shfl_xor(s, off, 32);
            if (lane == 0) sh_sc[wave] = s;
        }
        __syncthreads();

        const float invs = 1.f / fmaxf(sqrtf(sh_sc[0]), NORM_EPS_);
        const float inve = 1.f / fmaxf(sqrtf(sh_sc[1]), NORM_EPS_);

        {   // matvec: tid<256 -> Ms@kns, tid>=256 -> Me@kne (2 partials/row)
            const int  row  = tid & (HALF_ - 1);
            const int  part = (tid >> 7) & 1;
            const bool isE  = tid >= 256;
            const float* kk  = isE ? sh_ke : sh_ks;
            const float  inv = isE ? inve : invs;
            float (*M)[HALF_ + 1] = isE ? Me : Ms;
            float s = 0.f;
            const int j0 = part * 64;
            #pragma unroll 8
            for (int j = j0; j < j0 + 64; ++j) s += M[row][j] * (kk[j] * inv);
            sh_part[tid >> 7][row] = s;
        }
        __syncthreads();

        if (tid < HALF_) {
            sh_rs[tid] = sh_ks[tid] - (sh_part[0][tid] + sh_part[1][tid]);
        } else if (tid < 2 * HALF_) {
            const int r = tid - HALF_;
            sh_re[r] = sh_ke[r] - (sh_part[2][r] + sh_part[3][r]);
        }
        __syncthreads();

        if (wave < 2) {                       // residual norms
            const float* src = wave == 0 ? sh_rs : sh_re;
            float s = 0.f;
            #pragma unroll
            for (int i = lane; i < HALF_; i += 32) { const float x = src[i]; s += x * x; }
            #pragma unroll
            for (int off = 16; off > 0; off >>= 1) s += __shfl_xor(s, off, 32);
            if (lane == 0) sh_sc[2 + wave] = s;
        }
        __syncthreads();

        const bool fire = (sh_sc[2] >= GT2 * sh_sc[0]) ||
                          (sh_sc[3] >= GT2 * sh_sc[1]);
        if (fire) {                           // gated rank-1 updates
            const float wt  = (float)(t + 1) * (1.f / LL);
            const int   row = tid >> 2;       // 0..127
            const int   c0  = (tid & 3) * 32;
            const float rs  = sh_rs[row] * (1.f - ALPHA_);
            const float re  = sh_re[row] * (1.f - ALPHA_) * wt;
            #pragma unroll 8
            for (int j = 0; j < 32; ++j) {
                const int col = c0 + j;
                Ms[row][col] += rs * (sh_ks[col] * invs);
                Me[row][col] += re * (sh_ke[col] * inve);
            }
        }
        __syncthreads();
    }

    // readout: cs = Ms @ (q@Wsem) = Ms @ ks_all[:, L-1] (raw, not normalized)
    if (tid < HALF_) {
        const size_t o = ((size_t)b * LL + (LL - 1)) * HALF_;
        sh_ks[tid] = ks_all[o + tid];
        sh_ke[tid] = ke_all[o + tid];
    }
    __syncthreads();
    {
        const int  row  = tid & (HALF_ - 1);
        const int  part = (tid >> 7) & 1;
        const bool isE  = tid >= 256;
        const float* kk = isE ? sh_ke : sh_ks;
        float (*M)[HALF_ + 1] = isE ? Me : Ms;
        float s = 0.f;
        const int j0 = part * 64;
        #pragma unroll 8
        for (int j = j0; j < j0 + 64; ++j) s += M[row][j] * kk[j];
        sh_part[tid >> 7][row] = s;
    }
    __syncthreads();
    if (tid < HALF_)
        c_out[(size_t)b * HH + tid] = sh_part[0][tid] + sh_part[1][tid];
    else if (tid < 2 * HALF_)
        c_out[(size_t)b * HH + HALF_ + (tid - HALF_)] =
            sh_part[2][tid - HALF_] + sh_part[3][tid - HALF_];
}

// ---------------------------------------------------------------------------
// Kernel 4: r = c @ Wrp + brp   (8x256 @ 256x256, tiny)
// ---------------------------------------------------------------------------
__global__ __launch_bounds__(256) void k_rp(
    const float* __restrict__ c, const float* __restrict__ Wrp,
    const float* __restrict__ brp, float* __restrict__ r)
{
    __shared__ float sh_c[BB][HH];
    const int tid = threadIdx.x;
    for (int i = tid; i < BB * HH; i += 256) (&sh_c[0][0])[i] = c[i];
    __syncthreads();
    const int n = tid;
    float acc[BB] = {};
    for (int k = 0; k < HH; ++k) {
        const float w = Wrp[(size_t)k * HH + n];
        #pragma unroll
        for (int bb = 0; bb < BB; ++bb) acc[bb] += sh_c[bb][k] * w;
    }
    const float bias = brp[n];
    #pragma unroll
    for (int bb = 0; bb < BB; ++bb) r[(size_t)bb * HH + n] = acc[bb] + bias;
}

// ---------------------------------------------------------------------------
// Kernel 5: out = r @ Wout + bout. Bandwidth-bound (51.5 MB of Wout);
// each thread owns one vocab column, 8 FMAs per loaded word.
// ---------------------------------------------------------------------------
__global__ __launch_bounds__(256) void k_out(
    const float* __restrict__ r, const float* __restrict__ Wout,
    const float* __restrict__ bout, float* __restrict__ out)
{
    __shared__ float sh_r[BB][HH];
    const int tid = threadIdx.x;
    for (int i = tid; i < BB * HH; i += 256) (&sh_r[0][0])[i] = r[i];
    __syncthreads();

    const int v = blockIdx.x * 256 + tid;
    if (v >= VV) return;

    float acc[BB] = {};
    for (int k = 0; k < HH; ++k) {
        if ((k & 7) == 0 && k + 8 < HH)
            __builtin_prefetch(Wout + (size_t)(k + 8) * VV + v, 0, 1);
        const float w = Wout[(size_t)k * VV + v];
        #pragma unroll
        for (int bb = 0; bb < BB; ++bb) acc[bb] += sh_r[bb][k] * w;
    }
    const float bias = bout[v];
    #pragma unroll
    for (int bb = 0; bb < BB; ++bb)
        out[(size_t)bb * VV + v] = acc[bb] + bias;
}

// ---------------------------------------------------------------------------
extern "C" void kernel_launch(void* const* d_in, const int* in_sizes, int n_in,
                              void* d_out, int out_size, void* d_ws, size_t ws_size,
                              hipStream_t stream)
{
    const int*   seq   = (const int*)  d_in[0];
    const float* embed = (const float*)d_in[1];
    const float* W1    = (const float*)d_in[2];
    const float* b1    = (const float*)d_in[3];
    const float* W2    = (const float*)d_in[4];
    const float* b2    = (const float*)d_in[5];
    const float* gamma = (const float*)d_in[6];
    const float* beta  = (const float*)d_in[7];
    const float* Wsem  = (const float*)d_in[8];
    const float* Wepi  = (const float*)d_in[9];
    const float* Wrp   = (const float*)d_in[10];
    const float* brp   = (const float*)d_in[11];
    const float* Wout  = (const float*)d_in[12];
    const float* bout  = (const float*)d_in[13];
    float* out = (float*)d_out;

    char* ws = (char*)d_ws;
    float* h_ln   = (float*)(ws);                                  // 8 MB
    float* ks_all = (float*)(ws + (size_t)8  * 1024 * 1024);       // 4 MB
    float* ke_all = (float*)(ws + (size_t)12 * 1024 * 1024);       // 4 MB
    float* c_buf  = (float*)(ws + (size_t)16 * 1024 * 1024);       // 8 KB
    float* r_buf  = (float*)(ws + (size_t)16 * 1024 * 1024 + 8192);// 8 KB

    k_embed_ff_ln<<<BB * (LL / 16), 256, 0, stream>>>(
        seq, embed, W1, b1, W2, b2, gamma, beta, h_ln);
    k_proj<<<BB * (LL / 16), 256, 0, stream>>>(
        h_ln, Wsem, Wepi, ks_all, ke_all);
    k_scan<<<BB, 512, 0, stream>>>(ks_all, ke_all, c_buf);
    k_rp<<<1, 256, 0, stream>>>(c_buf, Wrp, brp, r_buf);
    k_out<<<(VV + 255) / 256, 256, 0, stream>>>(r_buf, Wout, bout, out);
}